// Info_NCE_24086176596036
// MI455X (gfx1250) — compile-verified
//
#include <hip/hip_runtime.h>
#include <hip/hip_bf16.h>

// InfoNCE loss: sum_i [ logsumexp_j(sim[i,j]/T) - sim[i,i]/T ],  T = 0.07
// N = 8192 (derived from in_sizes[0] = N*N).
//
// Stage 1 (bandwidth-bound, single streaming pass):
//   one wave32 per row, float4 (global_load_b128) coalesced loads,
//   online logsumexp in the base-2 domain using v_exp_f32 / v_log_f32,
//   wave reduction via __shfl_xor (wave32). Writes perRow[i] to d_ws.
// Stage 2 (deterministic scalar reduction, CDNA5 WMMA):
//   one wave, V_WMMA_F32_16X16X4_F32 with B = ones accumulates row-sums of
//   64 consecutive f32 values per instruction (full f32 precision),
//   128 chained accumulations, then one shfl fold. Fixed summation order
//   -> bit-deterministic result (no float atomics).

typedef __attribute__((ext_vector_type(2))) float v2f;
typedef __attribute__((ext_vector_type(8))) float v8f;

// Constants (double-precision derived, rounded once to f32):
//   K      = (1/0.07) * log2(e)  -> scale into base-2 exp domain
//   INV_T  = 1/0.07
//   LN2    = ln(2)
#define K_SCALE   20.609929155556618f
#define INV_T     14.285714285714286f
#define LN2_F     0.6931471805599453f

__device__ __forceinline__ float exp2_hw(float x) { return __builtin_amdgcn_exp2f(x); }
__device__ __forceinline__ float log2_hw(float x) { return __builtin_amdgcn_logf(x); }

__global__ __launch_bounds__(256) void infonce_rows(const float* __restrict__ sim,
                                                    float* __restrict__ perRow,
                                                    int n) {
    const int lane = threadIdx.x & 31;
    const int wave = threadIdx.x >> 5;
    const int row  = blockIdx.x * 8 + wave;
    if (row >= n) return;

    const float* __restrict__ rowp = sim + (size_t)row * (size_t)n;
    const float4* __restrict__ rp4 = reinterpret_cast<const float4*>(rowp);

    // Each lane handles n/128 float4s, coalesced: iter i -> lanes cover
    // 128 consecutive floats (512 B per wave per iteration).
    const int iters = n >> 7;

    // Online logsumexp in base-2 domain: track m2 = max(y), s = sum exp2(y - m2),
    // with y = sim * K_SCALE.
    float m2 = -1.0e30f;   // exp2(-1e30 - x) flushes to +0 -> safe start
    float ss = 0.0f;

#pragma unroll 4
    for (int i = 0; i < iters; ++i) {
        float4 v = rp4[i * 32 + lane];
        const float y0 = v.x * K_SCALE;
        const float y1 = v.y * K_SCALE;
        const float y2 = v.z * K_SCALE;
        const float y3 = v.w * K_SCALE;
        const float cmax = fmaxf(fmaxf(y0, y1), fmaxf(y2, y3));
        const float nm   = fmaxf(m2, cmax);
        ss = ss * exp2_hw(m2 - nm);          // rescale old partial (1.0 if no new max)
        m2 = nm;
        ss += exp2_hw(y0 - m2) + exp2_hw(y1 - m2)
            + exp2_hw(y2 - m2) + exp2_hw(y3 - m2);
    }

    // Wave32 tree merge of (m2, ss) pairs.
#pragma unroll
    for (int off = 16; off > 0; off >>= 1) {
        const float mo = __shfl_xor(m2, off);
        const float so = __shfl_xor(ss, off);
        const float nm = fmaxf(m2, mo);
        ss = ss * exp2_hw(m2 - nm) + so * exp2_hw(mo - nm);
        m2 = nm;
    }

    if (lane == 0) {
        const float diag = rowp[row];                       // sim[row, row]
        const float lse  = (m2 + log2_hw(ss)) * LN2_F;      // back to natural log
        perRow[row] = lse - diag * INV_T;
    }
}

// Stage 2: sum n floats with V_WMMA_F32_16X16X4_F32, B = ones.
// A layout (32-bit A 16x4, wave32): lane L<16 holds (M=L, K=0..1) in VGPR0..1,
// lane L>=16 holds (M=L-16, K=2..3).  With A[m][k] = buf[64*chunk + 4*m + k],
// per-lane byte-contiguous pair starts at off = 4*(L&15) + 2*(L>>4)  (even -> 8B aligned).
// D[m][j] += sum_k A[m][k]; after all chunks, total = sum_m D[m][j] for any j.
// C/D layout: VGPR r, lanes 0-15 -> (M=r, N=lane); lanes 16-31 -> (M=r+8, N=lane-16).
// So lane j partial = sum of its 8 accumulator VGPRs; total = lane j + lane j+16.
__global__ __launch_bounds__(32) void reduce_wmma(const float* __restrict__ perRow,
                                                  float* __restrict__ out,
                                                  int n) {
    const int lane = threadIdx.x;

    v2f b; b[0] = 1.0f; b[1] = 1.0f;   // B = all ones (layout-independent)
    v8f c = {};                        // accumulator starts at zero

    const int chunks = n >> 6;         // 64 values per WMMA
    const int off = ((lane & 15) << 2) + ((lane >> 4) << 1);

    for (int i = 0; i < chunks; ++i) {
        const float2 d = *reinterpret_cast<const float2*>(perRow + (i << 6) + off);
        v2f a; a[0] = d.x; a[1] = d.y;
        // 8 args: (neg_a, A, neg_b, B, c_mod, C, reuse_a, reuse_b)
        c = __builtin_amdgcn_wmma_f32_16x16x4_f32(
                /*neg_a=*/false, a, /*neg_b=*/false, b,
                /*c_mod=*/(short)0, c, /*reuse_a=*/false, /*reuse_b=*/false);
    }

    float lsum = c[0] + c[1] + c[2] + c[3] + c[4] + c[5] + c[6] + c[7];
    const float other = __shfl_xor(lsum, 16);
    if (lane == 0) out[0] = lsum + other;
}

extern "C" void kernel_launch(void* const* d_in, const int* in_sizes, int n_in,
                              void* d_out, int out_size, void* d_ws, size_t ws_size,
                              hipStream_t stream) {
    const float* sim = (const float*)d_in[0];
    float* out = (float*)d_out;
    float* per = (float*)d_ws;      // n floats of scratch (32 KB for n=8192)

    // n*n = in_sizes[0]; n is a power of two (8192 in the reference).
    const long long tot = (long long)in_sizes[0];
    int n = 1;
    while ((long long)n * (long long)n < tot) n <<= 1;

    const int wavesPerBlock = 8;                    // 256 threads = 8 wave32
    dim3 grid(n / wavesPerBlock), block(32 * wavesPerBlock);
    infonce_rows<<<grid, block, 0, stream>>>(sim, per, n);
    reduce_wmma<<<1, 32, 0, stream>>>(per, out, n);
}